// detector_14070312862253
// MI455X (gfx1250) — compile-verified
//
#include <hip/hip_runtime.h>
#include <hip/hip_bf16.h>
#include <float.h>

// Problem constants (match reference)
#define NB   32
#define NR   300
#define NC   37
#define NCM  36
#define ND   2048
#define AP   48          // A rows padded to 3 WMMA M-tiles
#define SORT_N 512
#define NMS_IOU_TH 0.4f
#define SCORE_THR  0.1f

typedef __attribute__((ext_vector_type(2)))  float        v2f;
typedef __attribute__((ext_vector_type(8)))  float        v8f;
typedef __attribute__((ext_vector_type(4)))  unsigned int v4u;
typedef __attribute__((ext_vector_type(8)))  int          v8i;
typedef __attribute__((ext_vector_type(4)))  int          v4i;

// ---------------------------------------------------------------------------
// Kernel 1: per-(image,class) box decode + sort + greedy NMS.
// One block per (b, c). 512 threads (16 waves), bitonic sort over 512 slots.
// ---------------------------------------------------------------------------
__global__ __launch_bounds__(SORT_N)
void det_sort_nms_kernel(const float* __restrict__ boxes,
                         const float* __restrict__ scores,
                         const float* __restrict__ bbox_pred,
                         const float* __restrict__ im_info,
                         float* __restrict__ out,        // dets + keepf regions
                         float* __restrict__ keep_orig)  // [B,AP,R] padded scratch
{
    __shared__ float skey[SORT_N];      // scores (sort key)
    __shared__ int   sidx[SORT_N];      // original roi index (payload)
    __shared__ float pbox[SORT_N * 4];  // decoded boxes, pre-sort order
    __shared__ float sbx [NR * 4];      // boxes in sorted order
    __shared__ float sarea[NR];
    __shared__ int   skeep[NR];

    const int t  = threadIdx.x;
    const int bc = blockIdx.x;          // (b * NCM + c)
    const int b  = bc / NCM;
    const int c  = bc % NCM;
    const int j  = c + 1;               // skip background class 0

    const float im_h      = im_info[b * 3 + 0];
    const float im_w      = im_info[b * 3 + 1];
    const float inv_scale = 1.0f / im_info[2];   // reference uses im_info[0, 2]

    // ---- decode boxes for this class ----
    if (t < NR) {
        const float x1 = boxes[((size_t)b * NR + t) * 4 + 0];
        const float y1 = boxes[((size_t)b * NR + t) * 4 + 1];
        const float x2 = boxes[((size_t)b * NR + t) * 4 + 2];
        const float y2 = boxes[((size_t)b * NR + t) * 4 + 3];
        const float* bp = bbox_pred + ((size_t)b * NR + t) * (4 * NC) + 4 * j;
        const float dx = bp[0] * 0.1f, dy = bp[1] * 0.1f;
        const float dw = bp[2] * 0.2f, dh = bp[3] * 0.2f;

        const float w  = x2 - x1 + 1.0f, h = y2 - y1 + 1.0f;
        const float cx = x1 + 0.5f * w,  cy = y1 + 0.5f * h;
        const float pcx = dx * w + cx,   pcy = dy * h + cy;
        const float pw  = expf(dw) * w,  ph  = expf(dh) * h;

        float px1 = pcx - 0.5f * pw, py1 = pcy - 0.5f * ph;
        float px2 = pcx + 0.5f * pw, py2 = pcy + 0.5f * ph;
        px1 = fminf(fmaxf(px1, 0.0f), im_w - 1.0f) * inv_scale;
        py1 = fminf(fmaxf(py1, 0.0f), im_h - 1.0f) * inv_scale;
        px2 = fminf(fmaxf(px2, 0.0f), im_w - 1.0f) * inv_scale;
        py2 = fminf(fmaxf(py2, 0.0f), im_h - 1.0f) * inv_scale;

        pbox[t * 4 + 0] = px1; pbox[t * 4 + 1] = py1;
        pbox[t * 4 + 2] = px2; pbox[t * 4 + 3] = py2;
        skey[t] = scores[((size_t)b * NR + t) * NC + j];
    } else {
        pbox[t * 4 + 0] = 0.f; pbox[t * 4 + 1] = 0.f;
        pbox[t * 4 + 2] = 0.f; pbox[t * 4 + 3] = 0.f;
        skey[t] = -FLT_MAX;                 // padding sorts to the end
    }
    sidx[t] = t;

    // ---- bitonic sort, descending by score ----
    for (int k = 2; k <= SORT_N; k <<= 1) {
        for (int jj = k >> 1; jj > 0; jj >>= 1) {
            __syncthreads();
            const int ixj = t ^ jj;
            if (ixj > t) {
                const float a = skey[t], bb2 = skey[ixj];
                if ((a < bb2) == ((t & k) == 0)) {   // descending overall
                    skey[t] = bb2; skey[ixj] = a;
                    const int ti = sidx[t]; sidx[t] = sidx[ixj]; sidx[ixj] = ti;
                }
            }
        }
    }
    __syncthreads();

    // ---- gather sorted boxes, areas, init keep = score > threshold ----
    if (t < NR) {
        const int o = sidx[t];
        const float x1 = pbox[o * 4 + 0], y1 = pbox[o * 4 + 1];
        const float x2 = pbox[o * 4 + 2], y2 = pbox[o * 4 + 3];
        sbx[t * 4 + 0] = x1; sbx[t * 4 + 1] = y1;
        sbx[t * 4 + 2] = x2; sbx[t * 4 + 3] = y2;
        sarea[t] = fmaxf(x2 - x1, 0.0f) * fmaxf(y2 - y1, 0.0f);
        skeep[t] = (skey[t] > SCORE_THR) ? 1 : 0;
    }

    // ---- greedy sequential NMS (thread t owns row t) ----
    for (int i = 0; i < NR - 1; ++i) {
        __syncthreads();
        if (t < NR && t > i && skeep[t] && skeep[i]) {
            const float ix1 = fmaxf(sbx[i * 4 + 0], sbx[t * 4 + 0]);
            const float iy1 = fmaxf(sbx[i * 4 + 1], sbx[t * 4 + 1]);
            const float ix2 = fminf(sbx[i * 4 + 2], sbx[t * 4 + 2]);
            const float iy2 = fminf(sbx[i * 4 + 3], sbx[t * 4 + 3]);
            const float inter = fmaxf(ix2 - ix1, 0.0f) * fmaxf(iy2 - iy1, 0.0f);
            const float iou = inter / (sarea[i] + sarea[t] - inter + 1e-8f);
            if (iou > NMS_IOU_TH) skeep[t] = 0;
        }
    }
    __syncthreads();

    // ---- class j==1 (c==0): keep only the top-scoring survivor ----
    if (c == 0 && t == 0) {
        int cnt = 0;
        for (int r = 0; r < NR; ++r)
            if (skeep[r]) { ++cnt; if (cnt > 1) skeep[r] = 0; }
    }
    __syncthreads();

    // ---- outputs: dets, keepf, and keep scattered back to roi order ----
    if (t < NR) {
        const float kf = skeep[t] ? 1.0f : 0.0f;
        const size_t base = (size_t)bc * NR + t;
        float* dets = out + base * 5;
        dets[0] = fmaxf(sbx[t * 4 + 0], 0.0f) * kf;
        dets[1] = fmaxf(sbx[t * 4 + 1], 0.0f) * kf;
        dets[2] = fmaxf(sbx[t * 4 + 2], 0.0f) * kf;
        dets[3] = fmaxf(sbx[t * 4 + 3], 0.0f) * kf;
        dets[4] = skey[t] * kf;
        out[(size_t)NB * NCM * NR * 5 + base] = kf;                 // keepf
        keep_orig[((size_t)b * AP + c) * NR + sidx[t]] = kf;        // padded scatter
    }
    // zero the 12 padding rows of this batch's A matrix (block c==0 owns them)
    if (c == 0 && t < NR) {
        for (int m = NCM; m < AP; ++m)
            keep_orig[((size_t)b * AP + m) * NR + t] = 0.0f;
    }
}

// ---------------------------------------------------------------------------
// Kernel 2: feat_sum[b] = keep_orig[b] (36x300, padded 48x300) @ feats[b] (300x2048)
// Block = 8 waves, one batch; A staged in LDS via TDM; each wave computes a
// 48x16 output strip (3 accumulators), so feats is read exactly once.
// ---------------------------------------------------------------------------
#define WAVES_PER_BLOCK 8
#define NT_TILES (ND / 16)                 // 128 N-tiles per batch
#define BLOCKS_PER_BATCH (NT_TILES / WAVES_PER_BLOCK)   // 16

__global__ __launch_bounds__(WAVES_PER_BLOCK * 32)
void feat_wmma_kernel(const float* __restrict__ keep_orig,
                      const float* __restrict__ feats,
                      float* __restrict__ outF)
{
    __shared__ __align__(16) float As[AP * NR];   // 48x300 f32 = 57.6 KB

    const int b        = blockIdx.x / BLOCKS_PER_BATCH;
    const int nt_group = blockIdx.x % BLOCKS_PER_BATCH;
    const int wave     = threadIdx.x >> 5;
    const int lane     = threadIdx.x & 31;
    const int nt       = nt_group * WAVES_PER_BLOCK + wave;

    const float* Ag = keep_orig + (size_t)b * AP * NR;

#if defined(__has_builtin) && __has_builtin(__builtin_amdgcn_tensor_load_to_lds)
    // Stage A via the Tensor Data Mover: 2D tile, 300 x 48 elements, 4B each.
    // (6-arg builtin on this toolchain: g0, g1, g2, g3, g4, cpol)
    if (wave == 0) {
        const unsigned long long gaddr = (unsigned long long)(const void*)Ag;
        const unsigned int lds_off = (unsigned int)(unsigned long long)(const void*)&As[0];
        v4u g0;
        g0[0] = 1u;                                        // count=1, flags 0
        g0[1] = lds_off;                                   // lds_addr
        g0[2] = (unsigned int)(gaddr & 0xFFFFFFFFu);       // global_addr[31:0]
        g0[3] = (unsigned int)((gaddr >> 32) & 0x01FFFFFFu) | (2u << 30); // [56:32] | type=2
        v8i g1;
        g1[0] = (2 << 16);                   // workgroup_mask=0, data_size=4B
        g1[1] = (int)((unsigned)NR << 16);   // tensor_dim0[15:0] in [31:16]
        g1[2] = (int)((unsigned)AP << 16);   // dim0 hi=0 | tensor_dim1[15:0]
        g1[3] = (int)((unsigned)NR << 16);   // dim1 hi=0 | tile_dim0=300
        g1[4] = AP;                          // tile_dim1=48 | tile_dim2=0
        g1[5] = NR;                          // tensor_dim0_stride[31:0]=300
        g1[6] = 0;                           // stride0 hi | stride1 lo (unused, 2D)
        g1[7] = 0;
        const v4i z4 = {0, 0, 0, 0};
        const v8i z8 = {0, 0, 0, 0, 0, 0, 0, 0};
        __builtin_amdgcn_tensor_load_to_lds(g0, g1, z4, z4, z8, 0);
        __builtin_amdgcn_s_wait_tensorcnt(0);
    }
    __syncthreads();
#else
    for (int i = threadIdx.x; i < AP * NR; i += WAVES_PER_BLOCK * 32)
        As[i] = Ag[i];
    __syncthreads();
#endif

    const int half_id = lane >> 4;     // 0: lanes 0-15, 1: lanes 16-31
    const int nl      = lane & 15;

    // B column pointer for this lane (N = nt*16 + nl), rows strided by ND
    const float* Bcol = feats + (size_t)b * NR * ND + (nt * 16 + nl);

    v8f acc0 = {}, acc1 = {}, acc2 = {};

    // fp32 WMMA 16x16x4: A lane (M=lane&15, K=2*half_id+{0,1}),
    //                    B lane (K=2*half_id+{0,1}, N=lane&15)
    for (int k0 = 0; k0 < NR; k0 += 4) {         // 300 % 4 == 0, no tail
        const int ka = k0 + 2 * half_id;
        v2f bv;
        bv.x = Bcol[(size_t)ka * ND];
        bv.y = Bcol[(size_t)(ka + 1) * ND];
        if (k0 + 68 < NR) __builtin_prefetch(&Bcol[(size_t)(k0 + 68) * ND], 0, 0);

        // A fragments from LDS: 8-byte aligned pair -> ds_load_b64
        const float2 f0 = *(const float2*)&As[(0 * 16 + nl) * NR + ka];
        const float2 f1 = *(const float2*)&As[(1 * 16 + nl) * NR + ka];
        const float2 f2 = *(const float2*)&As[(2 * 16 + nl) * NR + ka];
        v2f a0, a1, a2;
        a0.x = f0.x; a0.y = f0.y;
        a1.x = f1.x; a1.y = f1.y;
        a2.x = f2.x; a2.y = f2.y;

        acc0 = __builtin_amdgcn_wmma_f32_16x16x4_f32(
                   false, a0, false, bv, (short)0, acc0, false, false);
        acc1 = __builtin_amdgcn_wmma_f32_16x16x4_f32(
                   false, a1, false, bv, (short)0, acc1, false, false);
        acc2 = __builtin_amdgcn_wmma_f32_16x16x4_f32(
                   false, a2, false, bv, (short)0, acc2, false, false);
    }

    // C/D layout: VGPR i -> M = mbase + i + 8*half_id, N = nbase + (lane&15)
    const int ncol = nt * 16 + nl;
    #pragma unroll
    for (int i = 0; i < 8; ++i) {
        const int mr = half_id * 8 + i;
        outF[((size_t)b * NCM + (mr)) * ND + ncol]      = acc0[i];   // rows 0..15
        outF[((size_t)b * NCM + (16 + mr)) * ND + ncol] = acc1[i];   // rows 16..31
        const int m2 = 32 + mr;                                      // rows 32..47
        if (m2 < NCM)
            outF[((size_t)b * NCM + m2) * ND + ncol] = acc2[i];
    }
}

// ---------------------------------------------------------------------------
extern "C" void kernel_launch(void* const* d_in, const int* in_sizes, int n_in,
                              void* d_out, int out_size, void* d_ws, size_t ws_size,
                              hipStream_t stream)
{
    const float* boxes        = (const float*)d_in[0];
    const float* scores       = (const float*)d_in[1];
    const float* bbox_pred    = (const float*)d_in[2];
    const float* im_info      = (const float*)d_in[3];
    const float* roi_features = (const float*)d_in[4];
    float* out       = (float*)d_out;
    float* keep_orig = (float*)d_ws;     // B*AP*NR floats = 1.84 MB scratch

    det_sort_nms_kernel<<<NB * NCM, SORT_N, 0, stream>>>(
        boxes, scores, bbox_pred, im_info, out, keep_orig);

    float* outF = out + (size_t)NB * NCM * NR * 5 + (size_t)NB * NCM * NR;
    const int blocks = NB * BLOCKS_PER_BATCH;                 // 512
    feat_wmma_kernel<<<blocks, WAVES_PER_BLOCK * 32, 0, stream>>>(
        keep_orig, roi_features, outF);
}